// deepspeech_LocalDotAtten_38654705664186
// MI455X (gfx1250) — compile-verified
//
#include <hip/hip_runtime.h>
#include <stdint.h>

typedef __attribute__((ext_vector_type(2))) float v2f;
typedef __attribute__((ext_vector_type(8))) float v8f;
typedef __attribute__((ext_vector_type(4))) int   i4v;

#define BATCH 32
#define TLEN  1024
#define DIN   2048
#define EDIM  768
#define DOUT  29
#define WINR  5
#define KWIN  11
#define RING  16

// ---- CDNA5 async global->LDS path (guarded: falls back to reg staging) ----
#if defined(__HIP_DEVICE_COMPILE__) &&                                        \
    __has_builtin(__builtin_amdgcn_global_load_async_to_lds_b128) &&          \
    __has_builtin(__builtin_amdgcn_global_load_async_to_lds_b32) &&           \
    __has_builtin(__builtin_amdgcn_s_wait_asynccnt)
#define USE_ASYNC 1
#else
#define USE_ASYNC 0
#endif

#if USE_ASYNC
__device__ __forceinline__ void async_g2l_b128(const float* g, const float* l) {
  __builtin_amdgcn_global_load_async_to_lds_b128(
      (__attribute__((address_space(1))) i4v*)(uintptr_t)g,
      (__attribute__((address_space(3))) i4v*)(uint32_t)(uintptr_t)l, 0, 0);
}
__device__ __forceinline__ void async_g2l_b32(const float* g, const float* l) {
  __builtin_amdgcn_global_load_async_to_lds_b32(
      (__attribute__((address_space(1))) int*)(uintptr_t)g,
      (__attribute__((address_space(3))) int*)(uint32_t)(uintptr_t)l, 0, 0);
}
#endif

// ---------------------------------------------------------------------------
// Phase 1: x_e[m][n] = sum_k x[m][k] * W_in[n][k] + b_in[n]
// WMMA f32 16x16x4, block = 32(M) x 64(N), K in 32-chunks.
// Async double-buffered LDS staging (ASYNCcnt) when available.
// ---------------------------------------------------------------------------
#define MBLK   32
#define NBLK   64
#define KCHUNK 32
#define NCHUNKS (DIN / KCHUNK)

__global__ __launch_bounds__(256) void gemm_inproj(
    const float* __restrict__ x,    // [32768, 2048]
    const float* __restrict__ Win,  // [768, 2048]
    const float* __restrict__ bin,  // [768]
    float* __restrict__ xe)         // [32768, 768]
{
  const int tid  = threadIdx.x;
  const int m0   = blockIdx.y * MBLK;
  const int n0   = blockIdx.x * NBLK;
  const int wave = tid >> 5;
  const int lane = tid & 31;
  const int lm   = lane & 15;   // M (A) / N (B) index inside tile
  const int hi   = lane >> 4;   // half-wave selects K pair
  const int tm   = (wave >> 2) * 16;  // 2 waves in M
  const int tn   = (wave & 3) * 16;   // 4 waves in N

  v8f c = {};

#if USE_ASYNC
  // +4 pad: rows stay 16B-aligned for b128 dests; stride 36 mod 64 is
  // conflict-free for 16-lane column reads.
  __shared__ float As[2][MBLK][KCHUNK + 4];
  __shared__ float Bs[2][KCHUNK][NBLK + 1];

  auto stage = [&](int buf, int kc) {
    // A tile: 32x32 f32, one b128 per thread (coalesced 128B per 8 lanes)
    {
      int r = tid >> 3, kkq = (tid & 7) * 4;
      async_g2l_b128(&x[(size_t)(m0 + r) * DIN + kc + kkq], &As[buf][r][kkq]);
    }
    // B tile with transpose: Bs[kk][n] = Win[n0+n][kc+kk], b32 per element
    #pragma unroll
    for (int i = 0; i < 8; ++i) {
      int idx = tid + i * 256;
      int n = idx >> 5, kk = idx & 31;
      async_g2l_b32(&Win[(size_t)(n0 + n) * DIN + kc + kk], &Bs[buf][kk][n]);
    }
  };

  stage(0, 0);
  for (int it = 0; it < NCHUNKS; ++it) {
    const int buf = it & 1;
    __builtin_amdgcn_s_wait_asynccnt(0);  // this buf's copies landed
    __syncthreads();                      // ... and everyone left other buf
    if (it + 1 < NCHUNKS) stage(buf ^ 1, (it + 1) * KCHUNK);
    #pragma unroll
    for (int kk = 0; kk < KCHUNK; kk += 4) {
      v2f a, b;
      a.x = As[buf][lm][kk + 2 * hi];
      a.y = As[buf][lm][kk + 2 * hi + 1];
      b.x = Bs[buf][kk + 2 * hi][tn + lm];
      b.y = Bs[buf][kk + 2 * hi + 1][tn + lm];
      c = __builtin_amdgcn_wmma_f32_16x16x4_f32(
              false, a, false, b, (short)0, c, false, false);
    }
  }
#else
  __shared__ float As[MBLK][KCHUNK + 1];
  __shared__ float Bs[KCHUNK][NBLK + 1];

  for (int kc = 0; kc < DIN; kc += KCHUNK) {
    #pragma unroll
    for (int i = 0; i < 4; ++i) {
      int idx = tid + i * 256;
      int r = idx >> 5, kk = idx & 31;
      As[r][kk] = x[(size_t)(m0 + r) * DIN + kc + kk];
    }
    #pragma unroll
    for (int i = 0; i < 8; ++i) {
      int idx = tid + i * 256;
      int n = idx >> 5, kk = idx & 31;
      Bs[kk][n] = Win[(size_t)(n0 + n) * DIN + kc + kk];
    }
    __syncthreads();
    #pragma unroll
    for (int kk = 0; kk < KCHUNK; kk += 4) {
      v2f a, b;
      a.x = As[lm][kk + 2 * hi];
      a.y = As[lm][kk + 2 * hi + 1];
      b.x = Bs[kk + 2 * hi][tn + lm];
      b.y = Bs[kk + 2 * hi + 1][tn + lm];
      c = __builtin_amdgcn_wmma_f32_16x16x4_f32(
              false, a, false, b, (short)0, c, false, false);
    }
    __syncthreads();
  }
#endif

  // C/D layout: VGPR j -> M = j (lanes 0-15) or j+8 (lanes 16-31), N = lane%16
  const int n = n0 + tn + lm;
  const float bias = bin[n];
  #pragma unroll
  for (int j = 0; j < 8; ++j) {
    int m = m0 + tm + j + 8 * hi;
    xe[(size_t)m * EDIM + n] = c[j] + bias;
  }
}

// ---------------------------------------------------------------------------
// Phase 2: sequential local-attention scan. One workgroup per batch sample.
// 256 threads; thread owns E-channels {tid, tid+256, tid+512}.
// W_q / W_out in registers; 16-row key ring in LDS, async-prefetched 1 step
// ahead (row t+6 issued at step t, waited at step t+1).
// ---------------------------------------------------------------------------
__global__ __launch_bounds__(256, 1) void scan_attend(
    const float* __restrict__ xe,      // [32, 1024, 768]
    const int*   __restrict__ lengths, // [32]
    const float* __restrict__ Wq,      // [768, 29]
    const float* __restrict__ bq,      // [768]
    const float* __restrict__ Wout,    // [29, 768]
    const float* __restrict__ bout,    // [29]
    float* __restrict__ probs,         // [32, 1024, 29]
    float* __restrict__ wgts)          // [32, 1024, 11]
{
  __shared__ float ring[RING][EDIM];   // sliding key window (48 KB)
  __shared__ float q_s[32];            // decoder state q_t (29 used)
  __shared__ float sred[8][KWIN];      // per-wave score partials
  __shared__ float qred[8][DOUT];      // per-wave logit partials
  __shared__ float wts_s[KWIN];        // softmax weights

  const int b    = blockIdx.x;
  const int tid  = threadIdx.x;
  const int lane = tid & 31;
  const int wave = tid >> 5;
  const int len  = lengths[b];
  const float* xb = xe + (size_t)b * TLEN * EDIM;
  float* pb = probs + (size_t)b * TLEN * DOUT;
  float* wb = wgts  + (size_t)b * TLEN * KWIN;

  const int e0 = tid, e1 = tid + 256, e2 = tid + 512;

  // register-resident weights
  float wq0[DOUT], wq1[DOUT], wq2[DOUT];
  float wo0[DOUT], wo1[DOUT], wo2[DOUT];
  #pragma unroll
  for (int o = 0; o < DOUT; ++o) {
    wq0[o] = Wq[(size_t)e0 * DOUT + o];
    wq1[o] = Wq[(size_t)e1 * DOUT + o];
    wq2[o] = Wq[(size_t)e2 * DOUT + o];
    wo0[o] = Wout[(size_t)o * EDIM + e0];
    wo1[o] = Wout[(size_t)o * EDIM + e1];
    wo2[o] = Wout[(size_t)o * EDIM + e2];
  }
  const float bq0 = bq[e0], bq1 = bq[e1], bq2 = bq[e2];

  // init q0 = ones, last logit = 9
  if (tid < 32) q_s[tid] = (tid == DOUT - 1) ? 9.0f : 1.0f;

  // preload key rows 0..5 (window for t=0 is pos -5..5)
  #pragma unroll
  for (int pos = 0; pos < WINR + 1; ++pos) {
#if USE_ASYNC
    if (tid < EDIM / 4)
      async_g2l_b128(xb + (size_t)pos * EDIM + tid * 4, &ring[pos][tid * 4]);
#else
    ring[pos][e0] = xb[(size_t)pos * EDIM + e0];
    ring[pos][e1] = xb[(size_t)pos * EDIM + e1];
    ring[pos][e2] = xb[(size_t)pos * EDIM + e2];
#endif
  }
  __syncthreads();

  const float scale = 0.03608439182435161f;  // 1/sqrt(768)

  for (int t = 0; t < TLEN; ++t) {
    // ---- 1. query projection: qe = W_q @ q + b_q (register GEMV)
    float qe0 = bq0, qe1 = bq1, qe2 = bq2;
    #pragma unroll
    for (int o = 0; o < DOUT; ++o) {
      float q = q_s[o];
      qe0 += wq0[o] * q; qe1 += wq1[o] * q; qe2 += wq2[o] * q;
    }
#if USE_ASYNC
    __builtin_amdgcn_s_wait_asynccnt(0);  // row t+5 (issued at t-1) landed
#endif
    __syncthreads();  // B1

    // ---- prefetch key row t+6 (consumed at step t+1; slot aliases row t-10,
    //      which is outside the live window t-5..t+5 -> no WAR hazard)
    const int posn = t + WINR + 1;
    if (posn < TLEN) {
      const int sl = posn & (RING - 1);
#if USE_ASYNC
      if (tid < EDIM / 4)
        async_g2l_b128(xb + (size_t)posn * EDIM + tid * 4, &ring[sl][tid * 4]);
#else
      ring[sl][e0] = xb[(size_t)posn * EDIM + e0];
      ring[sl][e1] = xb[(size_t)posn * EDIM + e1];
      ring[sl][e2] = xb[(size_t)posn * EDIM + e2];
#endif
    }

    // ---- 2. masked attention scores (partials per thread, keys cached)
    float kv0[KWIN], kv1[KWIN], kv2[KWIN], sp[KWIN];
    #pragma unroll
    for (int k = 0; k < KWIN; ++k) {
      int pos = t + k - WINR;
      bool valid = (pos >= 0) && (pos < len);
      if (valid) {
        int sl = pos & (RING - 1);
        kv0[k] = ring[sl][e0]; kv1[k] = ring[sl][e1]; kv2[k] = ring[sl][e2];
      } else {
        kv0[k] = 0.f; kv1[k] = 0.f; kv2[k] = 0.f;
      }
      sp[k] = qe0 * kv0[k] + qe1 * kv1[k] + qe2 * kv2[k];
    }
    #pragma unroll
    for (int k = 0; k < KWIN; ++k) {
      #pragma unroll
      for (int off = 16; off > 0; off >>= 1)
        sp[k] += __shfl_xor(sp[k], off, 32);
    }
    if (lane == 0) {
      #pragma unroll
      for (int k = 0; k < KWIN; ++k) sred[wave][k] = sp[k];
    }
    __syncthreads();  // B2

    // ---- 3. masked softmax over the 11-wide window (single lane: tiny)
    if (tid == 0) {
      float sc[KWIN], ww[KWIN];
      float mx = -3.0e38f; int any = 0;
      #pragma unroll
      for (int k = 0; k < KWIN; ++k) {
        int pos = t + k - WINR;
        bool valid = (pos >= 0) && (pos < len);
        float s = 0.f;
        #pragma unroll
        for (int w = 0; w < 8; ++w) s += sred[w][k];
        s *= scale;
        sc[k] = valid ? s : -3.0e38f;
        if (valid) { mx = fmaxf(mx, s); any = 1; }
      }
      float ssum = 0.f;
      #pragma unroll
      for (int k = 0; k < KWIN; ++k) {
        float e = (sc[k] > -1.0e38f) ? __expf(sc[k] - mx) : 0.f;
        ww[k] = e; ssum += e;
      }
      float inv = any ? (1.0f / ssum) : 0.f;  // all-masked row -> weights 0
      #pragma unroll
      for (int k = 0; k < KWIN; ++k) {
        float v = ww[k] * inv;
        wts_s[k] = v;
        wb[(size_t)t * KWIN + k] = v;
      }
    }
    __syncthreads();  // B3

    // ---- 4. context: out = wts @ keys (keys still in regs)
    float o0 = 0.f, o1 = 0.f, o2 = 0.f;
    #pragma unroll
    for (int k = 0; k < KWIN; ++k) {
      float w = wts_s[k];
      o0 += w * kv0[k]; o1 += w * kv1[k]; o2 += w * kv2[k];
    }
    if (t >= len) { o0 = 0.f; o1 = 0.f; o2 = 0.f; }  // finished -> blank

    // ---- 5. output projection: q_next = W_out @ out + b_out
    float qp[DOUT];
    #pragma unroll
    for (int o = 0; o < DOUT; ++o)
      qp[o] = wo0[o] * o0 + wo1[o] * o1 + wo2[o] * o2;
    #pragma unroll
    for (int o = 0; o < DOUT; ++o) {
      #pragma unroll
      for (int off = 16; off > 0; off >>= 1)
        qp[o] += __shfl_xor(qp[o], off, 32);
    }
    if (lane == 0) {
      #pragma unroll
      for (int o = 0; o < DOUT; ++o) qred[wave][o] = qp[o];
    }
    __syncthreads();  // B4

    // ---- 6. combine, update q, emit softmax probs (wave 0)
    if (tid < 32) {
      float qn = 0.f;
      if (tid < DOUT) {
        qn = bout[tid];
        #pragma unroll
        for (int w = 0; w < 8; ++w) qn += qred[w][tid];
        q_s[tid] = qn;
      }
      float v = (tid < DOUT) ? qn : -3.0e38f;
      float m = v;
      #pragma unroll
      for (int off = 16; off > 0; off >>= 1)
        m = fmaxf(m, __shfl_xor(m, off, 32));
      float ex = (tid < DOUT) ? __expf(v - m) : 0.f;
      float s = ex;
      #pragma unroll
      for (int off = 16; off > 0; off >>= 1)
        s += __shfl_xor(s, off, 32);
      if (tid < DOUT)
        pb[(size_t)t * DOUT + tid] = (t < len) ? (ex / s) : 0.f;
    }
    __syncthreads();  // B5: q_s update visible to next iteration
  }
}

// ---------------------------------------------------------------------------
extern "C" void kernel_launch(void* const* d_in, const int* in_sizes, int n_in,
                              void* d_out, int out_size, void* d_ws, size_t ws_size,
                              hipStream_t stream) {
  const float* x    = (const float*)d_in[0];  // [32,1024,2048]
  const int*   lens = (const int*)  d_in[1];  // [32]
  const float* Win  = (const float*)d_in[2];  // [768,2048]
  const float* bin  = (const float*)d_in[3];  // [768]
  const float* Wq   = (const float*)d_in[4];  // [768,29]
  const float* bq   = (const float*)d_in[5];  // [768]
  const float* Wout = (const float*)d_in[6];  // [29,768]
  const float* bout = (const float*)d_in[7];  // [29]

  float* probs = (float*)d_out;                                  // [32,1024,29]
  float* wgts  = probs + (size_t)BATCH * TLEN * DOUT;            // [32,1024,11]
  float* xe    = (float*)d_ws;                                   // 96 MB scratch

  dim3 gg(EDIM / NBLK, (BATCH * TLEN) / MBLK);  // 12 x 1024 blocks
  hipLaunchKernelGGL(gemm_inproj, gg, dim3(256), 0, stream, x, Win, bin, xe);
  hipLaunchKernelGGL(scan_attend, dim3(BATCH), dim3(256), 0, stream,
                     xe, lens, Wq, bq, Wout, bout, probs, wgts);
}